// ai23_62886911148266
// MI455X (gfx1250) — compile-verified
//
#include <hip/hip_runtime.h>
#include <math.h>

namespace {

constexpr int   B  = 16;
constexpr int   C  = 23;
constexpr int   H  = 256;
constexpr int   W  = 256;
constexpr int   HW = H * W;          // 65536
constexpr float COVER = 32.0f;

// Native vector type: accepted by __builtin_nontemporal_{load,store},
// lowers to global_{load,store}_b128.
typedef float f32x4 __attribute__((ext_vector_type(4)));

// ---------------------------------------------------------------------------
// Pass 1: zero the accumulator grid (harness poisons d_out; we must re-zero
// every call). b128 stores, regular temporal hint: we WANT these lines
// resident in L2 so the scatter atomics hit L2, not HBM.
// ---------------------------------------------------------------------------
__global__ __launch_bounds__(256) void bev_zero(f32x4* __restrict__ out, int n4) {
    int i = blockIdx.x * blockDim.x + threadIdx.x;
    if (i < n4) {
        f32x4 z = {0.f, 0.f, 0.f, 0.f};
        out[i] = z;
    }
}

// ---------------------------------------------------------------------------
// Pass 2: per-pixel softmax over 23 channels + scatter-add into BEV cell.
// One thread per (b, h, w) pixel. Inputs are stream-once -> nontemporal
// loads keep L2 free for the atomic-hot output. Channel lines are 256 KB
// apart -> explicit global_prefetch_b8 per channel.
// ---------------------------------------------------------------------------
__global__ __launch_bounds__(256) void bev_scatter(const float* __restrict__ ptx,
                                                   const float* __restrict__ ptz,
                                                   const float* __restrict__ logits,
                                                   float* __restrict__ out) {
    int p  = blockIdx.x * blockDim.x + threadIdx.x;   // pixel id in [0, B*HW)
    int b  = p >> 16;                                 // HW == 2^16
    int hw = p & (HW - 1);

    float xv = __builtin_nontemporal_load(ptx + p);
    float zv = __builtin_nontemporal_load(ptz + p);

    // px = round((x + 32) * (W-1) / 64) ; pz = round(z * (1-H)/32 + (H-1))
    float pxf = (xv + COVER) * ((float)(W - 1) / (2.0f * COVER));
    float pzf = zv * ((float)(1 - H) / COVER) + (float)(H - 1);
    int px = __float2int_rn(pxf);   // round-to-nearest-even == jnp.round
    int pz = __float2int_rn(pzf);
    bool valid = (px >= 0) & (px < W) & (pz >= 0) & (pz < H);
    int tgt = valid ? (pz * W + px) : 0;

    const float* lp = logits + (size_t)b * C * HW + hw;

    // Warm the 23 strided cachelines (global_prefetch_b8, no counter cost).
#pragma unroll
    for (int c = 0; c < C; ++c) {
        __builtin_prefetch(lp + (size_t)c * HW, 0, 0);
    }

    // Max-subtracted softmax, all 23 logits held in registers.
    float l[C];
    float m = -INFINITY;
#pragma unroll
    for (int c = 0; c < C; ++c) {
        l[c] = __builtin_nontemporal_load(lp + (size_t)c * HW);
        m = fmaxf(m, l[c]);
    }
    float s = 0.f;
#pragma unroll
    for (int c = 0; c < C; ++c) {
        l[c] = __expf(l[c] - m);    // v_exp_f32
        s += l[c];
    }
    float inv = 1.0f / s;

    // Invalid points contribute exactly zero in the reference -> skip.
    if (!valid) return;

    // 23 no-return global_atomic_add_f32, all hitting the L2-resident grid.
    float* op = out + (size_t)b * C * HW + tgt;
#pragma unroll
    for (int c = 0; c < C; ++c) {
        atomicAdd(op + (size_t)c * HW, l[c] * inv);
    }
}

// ---------------------------------------------------------------------------
// Pass 3: clip to [0,1]. Read hits L2 (grid is resident); final store is the
// last touch -> nontemporal so it streams out without thrashing.
// ---------------------------------------------------------------------------
__global__ __launch_bounds__(256) void bev_clip(f32x4* __restrict__ out, int n4) {
    int i = blockIdx.x * blockDim.x + threadIdx.x;
    if (i >= n4) return;
    f32x4 v = out[i];
    v.x = fminf(fmaxf(v.x, 0.f), 1.f);
    v.y = fminf(fmaxf(v.y, 0.f), 1.f);
    v.z = fminf(fmaxf(v.z, 0.f), 1.f);
    v.w = fminf(fmaxf(v.w, 0.f), 1.f);
    __builtin_nontemporal_store(v, out + i);
}

} // anonymous namespace

extern "C" void kernel_launch(void* const* d_in, const int* in_sizes, int n_in,
                              void* d_out, int out_size, void* d_ws, size_t ws_size,
                              hipStream_t stream) {
    (void)in_sizes; (void)n_in; (void)d_ws; (void)ws_size;

    const float* ptx    = (const float*)d_in[0];   // pt_cloud_x  [B,1,H,W]
    const float* ptz    = (const float*)d_in[1];   // pt_cloud_z  [B,1,H,W]
    const float* logits = (const float*)d_in[2];   // semseg_logits [B,C,H,W]
    float* out = (float*)d_out;                    // [B,C,H,W] f32

    const int n  = out_size;         // 16*23*256*256 = 24,117,248 (mult of 4)
    const int n4 = n >> 2;

    const int T = 256;
    bev_zero<<<(n4 + T - 1) / T, T, 0, stream>>>((f32x4*)out, n4);

    const int npix = B * HW;         // 1,048,576
    bev_scatter<<<npix / T, T, 0, stream>>>(ptx, ptz, logits, out);

    bev_clip<<<(n4 + T - 1) / T, T, 0, stream>>>((f32x4*)out, n4);
}